// Distmult_43765716746754
// MI455X (gfx1250) — compile-verified
//
#include <hip/hip_runtime.h>

// DistMult: out[b] = sigmoid( sum_d h[b,d] * rel[rid[b],d] * t[b,d] )
// B = 262144, D = 512, fp32. Bandwidth-bound (~1 GB traffic -> ~44us floor
// at 23.3 TB/s). Row-reduction done in the matrix pipe via
// V_WMMA_F32_16X16X4_F32 with an all-ones B matrix (row-sum trick):
// one wave32 handles a tile of 16 rows, 128 k-steps of 4 columns.

typedef float v2f __attribute__((ext_vector_type(2)));
typedef float v4f __attribute__((ext_vector_type(4)));
typedef float v8f __attribute__((ext_vector_type(8)));

#define EMB_D 512
#define ROWS_PER_WAVE 16
#define WAVES_PER_BLOCK 8
#define ROWS_PER_BLOCK (ROWS_PER_WAVE * WAVES_PER_BLOCK)  // 128

__global__ __launch_bounds__(256) void distmult_wmma_kernel(
    const float* __restrict__ head,      // [B, D]
    const float* __restrict__ tail,      // [B, D]
    const int*   __restrict__ rel_ids,   // [B]
    const float* __restrict__ rel,       // [R, D]
    float*       __restrict__ out,       // [B]
    int nrows)
{
  const int lane = threadIdx.x & 31;
  const int wave = threadIdx.x >> 5;
  const int tile = blockIdx.x * WAVES_PER_BLOCK + wave;
  const int b0   = tile * ROWS_PER_WAVE;
  if (b0 >= nrows) return;  // uniform across the wave -> EXEC stays all-1s

  // A-matrix (16x4 f32) lane mapping: lane L<16 -> (M=L, K=0,1),
  // lane L+16 -> (M=L, K=2,3). Each lane streams float2's from its row.
  const int row  = lane & 15;
  const int half = lane >> 4;          // 0: K pair (0,1), 1: K pair (2,3)
  const int b    = b0 + row;

  const float* hp = head + (size_t)b * EMB_D + half * 2;
  const float* tp = tail + (size_t)b * EMB_D + half * 2;
  const int    ri = rel_ids[b];        // gather id (table lives in L2: 2 MB)
  const float* rp = rel + (size_t)ri * EMB_D + half * 2;

  // All-ones B matrix (4x16): layout-independent since every element is 1.
  v2f ones;
  ones.x = 1.0f;
  ones.y = 1.0f;

  // Two independent accumulator chains to relax XDL RAW latency.
  v8f acc0 = {};
  v8f acc1 = {};

  #pragma unroll 4
  for (int k = 0; k < EMB_D; k += 8) {
    // chain 0: columns k .. k+3
    v2f h0 = *(const v2f*)(hp + k);
    v2f r0 = *(const v2f*)(rp + k);
    v2f t0 = *(const v2f*)(tp + k);
    // chain 1: columns k+4 .. k+7
    v2f h1 = *(const v2f*)(hp + k + 4);
    v2f r1 = *(const v2f*)(rp + k + 4);
    v2f t1 = *(const v2f*)(tp + k + 4);

    v2f p0 = h0 * r0 * t0;
    v2f p1 = h1 * r1 * t1;

    // D = A(16x4) x Ones(4x16) + C : every column of D gets the row sum.
    acc0 = __builtin_amdgcn_wmma_f32_16x16x4_f32(
        /*neg_a=*/false, p0, /*neg_b=*/false, ones,
        /*c_mod=*/(short)0, acc0, /*reuse_a=*/false, /*reuse_b=*/false);
    acc1 = __builtin_amdgcn_wmma_f32_16x16x4_f32(
        /*neg_a=*/false, p1, /*neg_b=*/false, ones,
        /*c_mod=*/(short)0, acc1, /*reuse_a=*/false, /*reuse_b=*/false);
  }

  v8f acc = acc0 + acc1;

  // Sigmoid on all lanes (cheap); only lanes 0 and 16 hold N=0 columns we
  // store. C/D layout: lane 0 acc[i] = row i, lane 16 acc[i] = row 8+i.
  v8f s;
  #pragma unroll
  for (int i = 0; i < 8; ++i) {
    s[i] = 1.0f / (1.0f + __expf(-acc[i]));
  }

  v4f lo = __builtin_shufflevector(s, s, 0, 1, 2, 3);
  v4f hi = __builtin_shufflevector(s, s, 4, 5, 6, 7);
  if (lane == 0) {
    *(v4f*)(out + b0)     = lo;   // rows b0 .. b0+3
    *(v4f*)(out + b0 + 4) = hi;   // rows b0+4 .. b0+7
  } else if (lane == 16) {
    *(v4f*)(out + b0 + 8)  = lo;  // rows b0+8 .. b0+11
    *(v4f*)(out + b0 + 12) = hi;  // rows b0+12 .. b0+15
  }
}

extern "C" void kernel_launch(void* const* d_in, const int* in_sizes, int n_in,
                              void* d_out, int out_size, void* d_ws, size_t ws_size,
                              hipStream_t stream) {
  const float* head    = (const float*)d_in[0];
  const float* tail    = (const float*)d_in[1];
  const int*   rel_ids = (const int*)d_in[2];
  const float* rel     = (const float*)d_in[3];
  float*       out     = (float*)d_out;

  const int nrows  = in_sizes[0] / EMB_D;                     // B = 262144
  const int blocks = (nrows + ROWS_PER_BLOCK - 1) / ROWS_PER_BLOCK;  // 2048

  distmult_wmma_kernel<<<blocks, 256, 0, stream>>>(head, tail, rel_ids, rel,
                                                   out, nrows);
}